// FlowComposite_6648609374844
// MI455X (gfx1250) — compile-verified
//
#include <hip/hip_runtime.h>
#include <hip/hip_bf16.h>

// Problem constants from the reference: flows [B, T, 2, H, W] fp32.
#define FC_B 8
#define FC_T 12
#define FC_H 544
#define FC_W 960
#define FC_HW (FC_H * FC_W)

// Block covers 320 consecutive x positions (10 wave32s); grid = (W/320, H, B).
// Every thread is valid: no bounds check, no integer division anywhere.
#define FC_BLOCK 320

// One thread per output pixel (b, y, x). Runs the full T-step flow-composition
// recurrence privately:
//   px = x + u - 0.5 ; py = y + v - 0.5
//   (su, sv) = bilinear_zero_pad(flows[b, t], px, py)
//   u += su ; v += sv
// Branchless border handling: validity folded into bilinear weights, indices
// clamped, so all lanes execute identical code (no EXEC divergence).
__global__ __launch_bounds__(FC_BLOCK) void FlowComposite_6648609374844_kernel(
    const float* __restrict__ flows,  // [B, T, 2, H, W]
    float* __restrict__ out)          // [B, 2, H, W]
{
    const int x = blockIdx.x * FC_BLOCK + threadIdx.x;  // 0..959
    const int y = blockIdx.y;                           // 0..543
    const int b = blockIdx.z;                           // 0..7
    const int rem = y * FC_W + x;                       // offset within a plane

    const float gx = (float)x;
    const float gy = (float)y;

    float u = 0.0f;
    float v = 0.0f;

    // Base of this batch's T*2 planes.
    const float* __restrict__ fb = flows + (size_t)b * (size_t)(FC_T * 2 * FC_HW);

    for (int t = 0; t < FC_T; ++t) {
        const float* __restrict__ cu = fb + (size_t)t * (size_t)(2 * FC_HW); // u-plane
        const float* __restrict__ cv = cu + FC_HW;                           // v-plane

        // Warm the next timestep's planes near this pixel. Accumulated flow is a
        // short random walk, so step t+1 samples land close to (x, y).
        if (t + 1 < FC_T) {
            __builtin_prefetch(cu + 2 * FC_HW + rem, 0, 1); // -> global_prefetch_b8
            __builtin_prefetch(cv + 2 * FC_HW + rem, 0, 1);
        }

        const float px = gx + u - 0.5f;
        const float py = gy + v - 0.5f;

        const float fx0 = floorf(px);
        const float fy0 = floorf(py);
        const float wx1 = px - fx0;
        const float wy1 = py - fy0;
        const float wx0 = 1.0f - wx1;
        const float wy0 = 1.0f - wy1;

        const int x0 = (int)fx0;
        const int y0 = (int)fy0;
        const int x1 = x0 + 1;
        const int y1 = y0 + 1;

        // Zero-padding: a corner contributes only if inside the image.
        const float vx0 = (x0 >= 0 && x0 <= FC_W - 1) ? 1.0f : 0.0f;
        const float vx1 = (x1 >= 0 && x1 <= FC_W - 1) ? 1.0f : 0.0f;
        const float vy0 = (y0 >= 0 && y0 <= FC_H - 1) ? 1.0f : 0.0f;
        const float vy1 = (y1 >= 0 && y1 <= FC_H - 1) ? 1.0f : 0.0f;

        const float w00 = wx0 * wy0 * vx0 * vy0;
        const float w10 = wx1 * wy0 * vx1 * vy0;
        const float w01 = wx0 * wy1 * vx0 * vy1;
        const float w11 = wx1 * wy1 * vx1 * vy1;

        // Clamped indices (weight is already zero when out of range).
        const int x0c = min(max(x0, 0), FC_W - 1);
        const int x1c = min(max(x1, 0), FC_W - 1);
        const int y0c = min(max(y0, 0), FC_H - 1);
        const int y1c = min(max(y1, 0), FC_H - 1);

        const int o00 = y0c * FC_W + x0c;
        const int o10 = y0c * FC_W + x1c;
        const int o01 = y1c * FC_W + x0c;
        const int o11 = y1c * FC_W + x1c;

        // 8 independent gathers per step — issued together, one wait.
        const float u00 = cu[o00];
        const float u10 = cu[o10];
        const float u01 = cu[o01];
        const float u11 = cu[o11];
        const float v00 = cv[o00];
        const float v10 = cv[o10];
        const float v01 = cv[o01];
        const float v11 = cv[o11];

        const float su = u00 * w00 + u10 * w10 + u01 * w01 + u11 * w11;
        const float sv = v00 * w00 + v10 * w10 + v01 * w01 + v11 * w11;

        u += su;
        v += sv;
    }

    // Output layout [B, 2, H, W]: stack([u, v], axis=1). Coalesced, write-once
    // data -> non-temporal stores so we don't evict the hot flow planes from L2.
    float* __restrict__ ob = out + (size_t)b * (size_t)(2 * FC_HW);
    __builtin_nontemporal_store(u, ob + rem);
    __builtin_nontemporal_store(v, ob + FC_HW + rem);
}

extern "C" void kernel_launch(void* const* d_in, const int* in_sizes, int n_in,
                              void* d_out, int out_size, void* d_ws, size_t ws_size,
                              hipStream_t stream) {
    (void)in_sizes; (void)n_in; (void)d_ws; (void)ws_size; (void)out_size;

    const float* flows = (const float*)d_in[0];
    float* out = (float*)d_out;

    dim3 block(FC_BLOCK, 1, 1);                 // 320 = 10 wave32s
    dim3 grid(FC_W / FC_BLOCK, FC_H, FC_B);     // (3, 544, 8) — exact cover

    FlowComposite_6648609374844_kernel<<<grid, block, 0, stream>>>(flows, out);
}